// AdaptiveTrendExtractor_70377334112418
// MI455X (gfx1250) — compile-verified
//
#include <hip/hip_runtime.h>

typedef __attribute__((ext_vector_type(2))) float v2f;
typedef __attribute__((ext_vector_type(8))) float v8f;

#define LEN    720
#define ROWS   8
#define HALO   4
#define XPITCH (LEN + 2 * HALO)

__global__ __launch_bounds__(256) void adaptive_trend_fused(
    const float* __restrict__ x,
    const float* __restrict__ cw0, const float* __restrict__ cb0,
    const float* __restrict__ cw1, const float* __restrict__ cb1,
    const float* __restrict__ cw2, const float* __restrict__ cb2,
    const float* __restrict__ cw3, const float* __restrict__ cb3,
    const float* __restrict__ W1,  const float* __restrict__ b1,
    const float* __restrict__ W2,  const float* __restrict__ b2,
    float* __restrict__ out, int BN)
{
    extern __shared__ float smem[];
    float* xs   = smem;                      // ROWS * XPITCH     (x rows + halo)
    float* fsm  = xs  + ROWS * XPITCH;       // 4 * ROWS * LEN    (conv features)
    float* entL = fsm + 4 * ROWS * LEN;      // 16 * 4            (entropy A-matrix)
    float* hL   = entL + 16 * 4;             // 16 * 32           (hidden layer)
    float* wb   = hL  + 16 * 32;             // ROWS * 4          (blend weights)

    const int tid  = threadIdx.x;
    const int wave = tid >> 5;
    const int lane = tid & 31;
    const long long row = (long long)blockIdx.x * ROWS + wave;
    const bool active = row < (long long)BN;
    const float* xr = x + row * LEN;

    // ---- uniform conv params into registers ----
    float k3[3], k5[5], k7[7], k9[9];
#pragma unroll
    for (int t = 0; t < 3; ++t) k3[t] = cw0[t];
#pragma unroll
    for (int t = 0; t < 5; ++t) k5[t] = cw1[t];
#pragma unroll
    for (int t = 0; t < 7; ++t) k7[t] = cw2[t];
#pragma unroll
    for (int t = 0; t < 9; ++t) k9[t] = cw3[t];
    const float bb0 = cb0[0], bb1 = cb1[0], bb2 = cb2[0], bb3 = cb3[0];

    // ---- stage x row (+halo, zero padded) into LDS; zero pad A-matrix rows 8..15 ----
    for (int i = lane; i < XPITCH; i += 32) {
        int idx = i - HALO;
        xs[wave * XPITCH + i] = (active && idx >= 0 && idx < LEN) ? xr[idx] : 0.0f;
    }
    if (tid < (16 - ROWS) * 4) entL[ROWS * 4 + tid] = 0.0f;
    __syncthreads();

    // ---- phase 1: four convs in one pass, track per-scale max ----
    const float* xsW = xs + wave * XPITCH;
    float* f0 = fsm + (0 * ROWS + wave) * LEN;
    float* f1 = fsm + (1 * ROWS + wave) * LEN;
    float* f2 = fsm + (2 * ROWS + wave) * LEN;
    float* f3 = fsm + (3 * ROWS + wave) * LEN;

    float m0 = -3.402823466e38f, m1 = m0, m2 = m0, m3 = m0;
    for (int l = lane; l < LEN; l += 32) {
        float xv[9];
#pragma unroll
        for (int t = 0; t < 9; ++t) xv[t] = xsW[l + t];   // xs[i] = x[i-4]
        float a0 = bb0, a1 = bb1, a2 = bb2, a3 = bb3;
#pragma unroll
        for (int t = 0; t < 3; ++t) a0 = fmaf(k3[t], xv[t + 3], a0);
#pragma unroll
        for (int t = 0; t < 5; ++t) a1 = fmaf(k5[t], xv[t + 2], a1);
#pragma unroll
        for (int t = 0; t < 7; ++t) a2 = fmaf(k7[t], xv[t + 1], a2);
#pragma unroll
        for (int t = 0; t < 9; ++t) a3 = fmaf(k9[t], xv[t], a3);
        f0[l] = a0; f1[l] = a1; f2[l] = a2; f3[l] = a3;
        m0 = fmaxf(m0, a0); m1 = fmaxf(m1, a1);
        m2 = fmaxf(m2, a2); m3 = fmaxf(m3, a3);
    }
#pragma unroll
    for (int off = 16; off > 0; off >>= 1) {
        m0 = fmaxf(m0, __shfl_xor(m0, off, 32));
        m1 = fmaxf(m1, __shfl_xor(m1, off, 32));
        m2 = fmaxf(m2, __shfl_xor(m2, off, 32));
        m3 = fmaxf(m3, __shfl_xor(m3, off, 32));
    }

    // ---- phase 2: softmax denominators ----
    float s0 = 0.f, s1 = 0.f, s2 = 0.f, s3 = 0.f;
    for (int l = lane; l < LEN; l += 32) {
        s0 += __expf(f0[l] - m0); s1 += __expf(f1[l] - m1);
        s2 += __expf(f2[l] - m2); s3 += __expf(f3[l] - m3);
    }
#pragma unroll
    for (int off = 16; off > 0; off >>= 1) {
        s0 += __shfl_xor(s0, off, 32); s1 += __shfl_xor(s1, off, 32);
        s2 += __shfl_xor(s2, off, 32); s3 += __shfl_xor(s3, off, 32);
    }
    const float i0 = 1.f / s0, i1 = 1.f / s1, i2 = 1.f / s2, i3 = 1.f / s3;

    // ---- phase 3: entropy  ent = -sum p*log(p + 1e-8) ----
    float e0 = 0.f, e1 = 0.f, e2 = 0.f, e3 = 0.f;
    for (int l = lane; l < LEN; l += 32) {
        float p;
        p = __expf(f0[l] - m0) * i0; e0 = fmaf(-p, __logf(p + 1e-8f), e0);
        p = __expf(f1[l] - m1) * i1; e1 = fmaf(-p, __logf(p + 1e-8f), e1);
        p = __expf(f2[l] - m2) * i2; e2 = fmaf(-p, __logf(p + 1e-8f), e2);
        p = __expf(f3[l] - m3) * i3; e3 = fmaf(-p, __logf(p + 1e-8f), e3);
    }
#pragma unroll
    for (int off = 16; off > 0; off >>= 1) {
        e0 += __shfl_xor(e0, off, 32); e1 += __shfl_xor(e1, off, 32);
        e2 += __shfl_xor(e2, off, 32); e3 += __shfl_xor(e3, off, 32);
    }
    if (lane == 0) {
        entL[wave * 4 + 0] = active ? e0 : 0.f;
        entL[wave * 4 + 1] = active ? e1 : 0.f;
        entL[wave * 4 + 2] = active ? e2 : 0.f;
        entL[wave * 4 + 3] = active ? e3 : 0.f;
    }
    __syncthreads();

    // ---- phase 4: layer 1 via V_WMMA_F32_16X16X4_F32  (h = ent(16x4) @ W1(4x32)) ----
    if (wave == 0) {
        const int M  = lane & 15;               // A/D row within half
        const int kh = (lane < 16) ? 0 : 2;     // K pair held by this lane half
        const int nl = lane & 15;               // B/D column within half

        // A layout (ISA 16x4 f32): V0 = K{0|2}, V1 = K{1|3}, M = lane&15
        v2f A;
        A.x = entL[M * 4 + kh];
        A.y = entL[M * 4 + kh + 1];

        // B (4x16 slices of W1, row-major W1[k*32+n]); mirror of A layout with N on lanes
        v2f B0, B1;
        B0.x = W1[(kh    ) * 32 + nl];       B0.y = W1[(kh + 1) * 32 + nl];
        B1.x = W1[(kh    ) * 32 + 16 + nl];  B1.y = W1[(kh + 1) * 32 + 16 + nl];

        v8f z = {0.f, 0.f, 0.f, 0.f, 0.f, 0.f, 0.f, 0.f};
        v8f d0 = __builtin_amdgcn_wmma_f32_16x16x4_f32(false, A, false, B0,
                                                       (short)0, z, false, false);
        v8f d1 = __builtin_amdgcn_wmma_f32_16x16x4_f32(false, A, false, B1,
                                                       (short)0, z, false, false);

        // D layout: VGPR r -> rows r (lanes 0-15) and r+8 (lanes 16-31)
#pragma unroll
        for (int r = 0; r < 8; ++r) {
            int hr = (lane < 16) ? r : r + 8;
            hL[hr * 32 + nl]      = fmaxf(d0[r] + b1[nl],      0.0f);
            hL[hr * 32 + 16 + nl] = fmaxf(d1[r] + b1[16 + nl], 0.0f);
        }
    }
    __syncthreads();

    // ---- phase 5: layer 2 (32->4) + softmax -> blend weights ----
    if (tid < ROWS) {
        float g0 = b2[0], g1 = b2[1], g2 = b2[2], g3 = b2[3];
#pragma unroll
        for (int i = 0; i < 32; ++i) {
            float hv = hL[tid * 32 + i];
            g0 = fmaf(hv, W2[i * 4 + 0], g0);
            g1 = fmaf(hv, W2[i * 4 + 1], g1);
            g2 = fmaf(hv, W2[i * 4 + 2], g2);
            g3 = fmaf(hv, W2[i * 4 + 3], g3);
        }
        float gm = fmaxf(fmaxf(g0, g1), fmaxf(g2, g3));
        float x0 = __expf(g0 - gm), x1 = __expf(g1 - gm);
        float x2 = __expf(g2 - gm), x3 = __expf(g3 - gm);
        float inv = 1.f / (x0 + x1 + x2 + x3);
        wb[tid * 4 + 0] = x0 * inv; wb[tid * 4 + 1] = x1 * inv;
        wb[tid * 4 + 2] = x2 * inv; wb[tid * 4 + 3] = x3 * inv;
    }
    __syncthreads();

    // ---- phase 6: weighted blend, single coalesced store pass ----
    const float q0 = wb[wave * 4 + 0], q1 = wb[wave * 4 + 1];
    const float q2 = wb[wave * 4 + 2], q3 = wb[wave * 4 + 3];
    float* orow = out + row * LEN;
    for (int l = lane; l < LEN; l += 32) {
        float r = q0 * f0[l] + q1 * f1[l] + q2 * f2[l] + q3 * f3[l];
        if (active) orow[l] = r;
    }
}

extern "C" void kernel_launch(void* const* d_in, const int* in_sizes, int n_in,
                              void* d_out, int out_size, void* d_ws, size_t ws_size,
                              hipStream_t stream) {
    (void)n_in; (void)out_size; (void)d_ws; (void)ws_size;
    const float* x   = (const float*)d_in[0];
    const float* cw0 = (const float*)d_in[1];  const float* cb0 = (const float*)d_in[2];
    const float* cw1 = (const float*)d_in[3];  const float* cb1 = (const float*)d_in[4];
    const float* cw2 = (const float*)d_in[5];  const float* cb2 = (const float*)d_in[6];
    const float* cw3 = (const float*)d_in[7];  const float* cb3 = (const float*)d_in[8];
    const float* W1  = (const float*)d_in[9];  const float* b1  = (const float*)d_in[10];
    const float* W2  = (const float*)d_in[11]; const float* b2  = (const float*)d_in[12];
    float* out = (float*)d_out;

    const int BN = in_sizes[0] / LEN;                    // 64*321 = 20544
    const int blocks = (BN + ROWS - 1) / ROWS;           // 2568
    const size_t shmem = (size_t)(ROWS * XPITCH + 4 * ROWS * LEN
                                  + 16 * 4 + 16 * 32 + ROWS * 4) * sizeof(float);

    adaptive_trend_fused<<<dim3(blocks), dim3(256), shmem, stream>>>(
        x, cw0, cb0, cw1, cb1, cw2, cb2, cw3, cb3, W1, b1, W2, b2, out, BN);
}